// EmbeddingModel_Optimized_56160992362527
// MI455X (gfx1250) — compile-verified
//
#include <hip/hip_runtime.h>

#define ENT_SIZE 250000
#define REL_MAX  63
#define EMBED    128
#define BATCH    32768
#define KNEG     8

typedef __attribute__((ext_vector_type(2))) float v2f;
typedef __attribute__((ext_vector_type(8))) float v8f;

__device__ __forceinline__ float wave_sum(float v) {
#pragma unroll
    for (int off = 16; off > 0; off >>= 1)
        v += __shfl_xor(v, off, 32);
    return v;
}

__device__ __forceinline__ float logsig(float x) {
    // numerically stable log(sigmoid(x)) = min(x,0) - log1p(exp(-|x|))
    return fminf(x, 0.0f) - log1pf(__expf(-fabsf(x)));
}

__device__ __forceinline__ int branch_of(int il, int pl) {
    return ((il >= ENT_SIZE) ? 2 : 0) + ((pl >= ENT_SIZE) ? 1 : 0);
}

// ---------------- Stage 1: per-block branch histograms ----------------
__global__ void hist_kernel(const int* __restrict__ i_labels,
                            const int* __restrict__ p_labels,
                            int i_shift, int* __restrict__ g_hist) {
    __shared__ int h[4];
    int t = threadIdx.x;
    if (t < 4) h[t] = 0;
    __syncthreads();
    int row = blockIdx.x * 256 + t;
    int il = i_labels[row >> i_shift];
    int pl = p_labels[row];
    atomicAdd(&h[branch_of(il, pl)], 1);
    __syncthreads();
    if (t < 4) g_hist[blockIdx.x * 4 + t] = h[t];
}

// ---------------- Stage 2: scan histograms -> per-block bases ----------------
__global__ void scan_kernel(int* __restrict__ g_hist, int nblocks) {
    __shared__ int totals[4];
    __shared__ int base[4];
    int t = threadIdx.x;
    if (t < 4) {
        int run = 0;
        for (int b = 0; b < nblocks; ++b) {
            int c = g_hist[b * 4 + t];
            g_hist[b * 4 + t] = run;
            run += c;
        }
        totals[t] = run;
    }
    __syncthreads();
    if (t == 0) {
        int run = 0;
        for (int v = 0; v < 4; ++v) { base[v] = run; run += totals[v]; }
    }
    __syncthreads();
    if (t < 4)
        for (int b = 0; b < nblocks; ++b) g_hist[b * 4 + t] += base[t];
}

// ---------------- Stage 3: stable rank -> destination index ----------------
__global__ void rank_kernel(const int* __restrict__ i_labels,
                            const int* __restrict__ p_labels,
                            int i_shift, const int* __restrict__ g_hist,
                            int* __restrict__ dest_idx) {
    __shared__ int wave_hist[8][4];
    __shared__ int wave_pref[8][4];
    int t = threadIdx.x;
    int wave = t >> 5, lane = t & 31;
    int row = blockIdx.x * 256 + t;
    int il = i_labels[row >> i_shift];
    int pl = p_labels[row];
    int br = branch_of(il, pl);
    unsigned b0 = __builtin_amdgcn_ballot_w32((br & 1) != 0);
    unsigned b1 = __builtin_amdgcn_ballot_w32((br & 2) != 0);
    unsigned same = ((br & 1) ? b0 : ~b0) & ((br & 2) ? b1 : ~b1);
    unsigned ltmask = (1u << lane) - 1u;
    int rank = __popc(same & ltmask);
    int cnt  = __popc(same);
    if (lane < 4) wave_hist[wave][lane] = 0;
    __syncthreads();
    if ((same & ltmask) == 0u) wave_hist[wave][br] = cnt;  // lowest lane of group
    __syncthreads();
    if (t < 4) {
        int run = 0;
        for (int w = 0; w < 8; ++w) {
            wave_pref[w][t] = run;
            run += wave_hist[w][t];
        }
    }
    __syncthreads();
    dest_idx[row] = g_hist[blockIdx.x * 4 + br] + wave_pref[wave][br] + rank;
}

// ---------------- Stage 4: gather + project + WMMA dot + scatter ----------------
// 2 waves per block, 16 rows per wave. One V_WMMA_F32_16X16X4_F32 chain per wave.
__global__ void pair_dot_kernel(const float* __restrict__ in_ent,
                                const float* __restrict__ out_ent,
                                const float* __restrict__ in_rel,
                                const float* __restrict__ out_rel,
                                const float* __restrict__ in_map,
                                const float* __restrict__ out_map,
                                const int* __restrict__ i_labels,
                                const int* __restrict__ p_labels,
                                const int* __restrict__ dest_idx,
                                float* __restrict__ sorted_vals,
                                int i_shift, float sign) {
    __shared__ float lds_inp[2][16 * EMBED];
    __shared__ float lds_out[2][16 * EMBED];
    int wave = threadIdx.x >> 5;
    int lane = threadIdx.x & 31;
    int row0 = (blockIdx.x * 2 + wave) * 16;
    int kk = lane * 4;

    // ---- Phase 1: build inp/out rows into LDS (branches are wave-uniform) ----
    for (int r = 0; r < 16; ++r) {
        int row = row0 + r;
        int il = i_labels[row >> i_shift];
        int pl = p_labels[row];
        bool ei = il < ENT_SIZE;
        bool ep = pl < ENT_SIZE;
        int iid = ei ? il : il - ENT_SIZE;
        int pid = ep ? pl : pl - ENT_SIZE;
        int iid_r = iid < REL_MAX ? iid : REL_MAX;
        int pid_r = pid < REL_MAX ? pid : REL_MAX;

        float4 inp, outp;
        // input embedding
        if (ei) {
            float4 e = *(const float4*)(in_ent + (size_t)iid * EMBED + kk);
            if (ep) {
                inp = e;
            } else {
                float4 m = *(const float4*)(in_map + (size_t)pid_r * EMBED + kk);
                float ss = wave_sum(m.x * m.x + m.y * m.y + m.z * m.z + m.w * m.w);
                float em = wave_sum(e.x * m.x + e.y * m.y + e.z * m.z + e.w * m.w);
                float nrm = fmaxf(sqrtf(ss), 1e-12f);
                float coef = em / (nrm * nrm);
                inp.x = e.x - coef * m.x; inp.y = e.y - coef * m.y;
                inp.z = e.z - coef * m.z; inp.w = e.w - coef * m.w;
            }
        } else {
            inp = *(const float4*)(in_rel + (size_t)iid_r * EMBED + kk);
        }
        // output embedding
        if (ep) {
            float4 e = *(const float4*)(out_ent + (size_t)pid * EMBED + kk);
            if (ei) {
                outp = e;
            } else {
                float4 m = *(const float4*)(out_map + (size_t)iid_r * EMBED + kk);
                float ss = wave_sum(m.x * m.x + m.y * m.y + m.z * m.z + m.w * m.w);
                float em = wave_sum(e.x * m.x + e.y * m.y + e.z * m.z + e.w * m.w);
                float nrm = fmaxf(sqrtf(ss), 1e-12f);
                float coef = em / (nrm * nrm);
                outp.x = e.x - coef * m.x; outp.y = e.y - coef * m.y;
                outp.z = e.z - coef * m.z; outp.w = e.w - coef * m.w;
            }
        } else {
            outp = *(const float4*)(out_rel + (size_t)pid_r * EMBED + kk);
        }
        *(float4*)(&lds_inp[wave][r * EMBED + kk]) = inp;
        *(float4*)(&lds_out[wave][r * EMBED + kk]) = outp;
    }

    // ---- Phase 2: 16x16 cross-dot via V_WMMA_F32_16X16X4_F32, K = 128 ----
    // A 16x4 layout: lane l holds row (l&15), K = 4*kc + (l>=16 ? 2:0) + {0,1}
    // B 4x16 layout: lane l holds col (l&15) (= out row), same K pattern.
    int rsel = lane & 15;
    int koff = (lane >> 4) ? 2 : 0;
    const float* ip = &lds_inp[wave][rsel * EMBED];
    const float* op = &lds_out[wave][rsel * EMBED];
    v8f acc = {};
#pragma unroll
    for (int kc = 0; kc < 32; ++kc) {
        int k = kc * 4 + koff;
        v2f a = *(const v2f*)(ip + k);
        v2f b = *(const v2f*)(op + k);
        acc = __builtin_amdgcn_wmma_f32_16x16x4_f32(
            /*neg_a=*/false, a, /*neg_b=*/false, b,
            /*c_mod=*/(short)0, acc, /*reuse_a=*/false, /*reuse_b=*/false);
    }

    // ---- Phase 3: diagonal extraction -> log-sigmoid -> scatter ----
    // D element (m,m): VGPR m%8, lane m (m<8) or lane m+16 (m>=8).
    float d = 0.0f;
#pragma unroll
    for (int v = 0; v < 8; ++v) {
        float lo = __shfl(acc[v], v, 32);
        float hi = __shfl(acc[v], v + 24, 32);
        if (lane == v) d = lo;
        if (lane == v + 8) d = hi;
    }
    if (lane < 16) {
        int row = row0 + lane;
        sorted_vals[dest_idx[row]] = logsig(sign * d);
    }
}

// ---------------- Stage 5: combine ----------------
__global__ void combine_kernel(const float* __restrict__ pos_sorted,
                               const float* __restrict__ neg_sorted,
                               float* __restrict__ out) {
    int j = blockIdx.x * 256 + threadIdx.x;
    float s = pos_sorted[j];
    float ns = 0.0f;
#pragma unroll
    for (int k = 0; k < KNEG; ++k) ns += neg_sorted[j * KNEG + k];
    out[j] = -(s + ns);
}

extern "C" void kernel_launch(void* const* d_in, const int* in_sizes, int n_in,
                              void* d_out, int out_size, void* d_ws, size_t ws_size,
                              hipStream_t stream) {
    (void)in_sizes; (void)n_in; (void)out_size; (void)ws_size;
    const float* in_ent  = (const float*)d_in[0];
    const float* out_ent = (const float*)d_in[1];
    const float* in_rel  = (const float*)d_in[2];
    const float* out_rel = (const float*)d_in[3];
    const float* in_map  = (const float*)d_in[4];
    const float* out_map = (const float*)d_in[5];
    const int* input_labels = (const int*)d_in[6];
    const int* pos_labels   = (const int*)d_in[7];
    const int* neg_labels   = (const int*)d_in[8];  // flat B*K
    float* out = (float*)d_out;

    const int N1 = BATCH;          // positive rows
    const int N2 = BATCH * KNEG;   // negative rows
    char* ws = (char*)d_ws;
    int*   pos_dest   = (int*)ws;            ws += (size_t)N1 * sizeof(int);
    int*   neg_dest   = (int*)ws;            ws += (size_t)N2 * sizeof(int);
    float* pos_sorted = (float*)ws;          ws += (size_t)N1 * sizeof(float);
    float* neg_sorted = (float*)ws;          ws += (size_t)N2 * sizeof(float);
    int*   pos_hist   = (int*)ws;            ws += (size_t)(N1 / 256) * 4 * sizeof(int);
    int*   neg_hist   = (int*)ws;

    hist_kernel<<<N1 / 256, 256, 0, stream>>>(input_labels, pos_labels, 0, pos_hist);
    hist_kernel<<<N2 / 256, 256, 0, stream>>>(input_labels, neg_labels, 3, neg_hist);
    scan_kernel<<<1, 32, 0, stream>>>(pos_hist, N1 / 256);
    scan_kernel<<<1, 32, 0, stream>>>(neg_hist, N2 / 256);
    rank_kernel<<<N1 / 256, 256, 0, stream>>>(input_labels, pos_labels, 0, pos_hist, pos_dest);
    rank_kernel<<<N2 / 256, 256, 0, stream>>>(input_labels, neg_labels, 3, neg_hist, neg_dest);
    pair_dot_kernel<<<N1 / 32, 64, 0, stream>>>(in_ent, out_ent, in_rel, out_rel,
                                                in_map, out_map, input_labels, pos_labels,
                                                pos_dest, pos_sorted, 0, 1.0f);
    pair_dot_kernel<<<N2 / 32, 64, 0, stream>>>(in_ent, out_ent, in_rel, out_rel,
                                                in_map, out_map, input_labels, neg_labels,
                                                neg_dest, neg_sorted, 3, -1.0f);
    combine_kernel<<<BATCH / 256, 256, 0, stream>>>(pos_sorted, neg_sorted, out);
}